// ROPE_64252710748794
// MI455X (gfx1250) — compile-verified
//
#include <hip/hip_runtime.h>
#include <math.h>

// ---------------------------------------------------------------------------
// MI455X (gfx1250) notes:
//  * Workload is a bandwidth-bound streaming dot-reduction: 128 MB read,
//    scalar out.  HBM roofline ~5.5us; fits in 192MB L2 across graph replays.
//  * Main loop: float4 (global_load_b128) grid-stride, per-lane f32 accum.
//  * Block reduction uses V_WMMA_F32_16X16X4_F32 as an f32 horizontal-add
//    engine: A (16x4, documented layout: 2 VGPRs/lane) holds 64 folded
//    partials, B = all-ones (layout-independent), D rows = row-sums.
//    Per-lane sum of D's 8 VGPRs + shfl_xor(16) completes the reduction.
//  * Deterministic: fixed grid, two-phase reduce via d_ws, no float atomics.
// ---------------------------------------------------------------------------

typedef __attribute__((ext_vector_type(2))) float v2f;
typedef __attribute__((ext_vector_type(8))) float v8f;

#define PI_F     3.14159265358979323846f
#define TWO_PI_F 6.28318530717958647692f

__device__ __forceinline__ float cordic_est(float theta) {
    // radians(deg) constants, computed in double, rounded once to f32
    const float DEG_3_5    = (float)(3.5    * 3.14159265358979323846 / 180.0);
    const float DEG_7_125  = (float)(7.125  * 3.14159265358979323846 / 180.0);
    const float DEG_8_13   = (float)(8.13   * 3.14159265358979323846 / 180.0);   // 16.26/2
    const float DEG_26_565 = (float)(26.565 * 3.14159265358979323846 / 180.0);   // (36.87+16.26)/2
    const float DEG_16_26  = (float)(16.26  * 3.14159265358979323846 / 180.0);
    const float DEG_36_87  = (float)(36.87  * 3.14159265358979323846 / 180.0);

    // trivial_rotation: jnp.mod(theta, 2pi); theta > 0 here so fmodf matches
    float t = fmodf(theta, TWO_PI_F);
    float sg = (t > 0.0f) ? 1.0f : -1.0f;          // _sgn: sgn(0) == -1
    t = (fabsf(t) > PI_F) ? (t - sg * TWO_PI_F) : t;
    float at = fabsf(t);
    float rotmag = (at > 0.75f * PI_F) ? PI_F
                 : ((at > 0.25f * PI_F) ? (0.5f * PI_F) : 0.0f);
    sg = (t > 0.0f) ? 1.0f : -1.0f;
    float r = t - sg * rotmag;

    // friend_angles
    float s  = (r > 0.0f) ? 1.0f : -1.0f;
    float ar = fabsf(r);
    r = (ar < DEG_8_13) ? r
      : ((ar < DEG_26_565) ? (r - s * DEG_16_26) : (r - s * DEG_36_87));

    // USRcordic micro-rotation
    ar = fabsf(r);
    float step = (r > 0.0f) ? -DEG_7_125 : DEG_7_125;
    r = (ar < DEG_3_5) ? r : (r + step);

    // cordicII_ang == theta - r ; est = 1.57 * sin(ang)
    return 1.57f * sinf(theta - r);
}

// Block-wide (256 threads) sum using one V_WMMA_F32_16X16X4_F32.
// Returns the full block sum in every thread.
__device__ __forceinline__ float block_reduce_wmma(float acc) {
    __shared__ float red[256];
    const int tid = threadIdx.x;
    red[tid] = acc;
    __syncthreads();

    // Fold 256 -> 64 and pack the 64 values into the documented A(16x4,f32)
    // layout: lane L(0..15): {A[L][0], A[L][1]}, lane 16+L: {A[L][2], A[L][3]}.
    // Every wave computes identical operands -> EXEC all-1s for the WMMA.
    const int l = tid & 31;
    float a0 = red[2 * l]     + red[2 * l + 64] + red[2 * l + 128] + red[2 * l + 192];
    float a1 = red[2 * l + 1] + red[2 * l + 65] + red[2 * l + 129] + red[2 * l + 193];

    v2f A; A.x = a0;   A.y = a1;
    v2f B; B.x = 1.0f; B.y = 1.0f;   // all-ones B: correct under any B layout
    v8f C = {};
    // (neg_a, A, neg_b, B, c_mod, C, reuse_a, reuse_b)
    v8f D = __builtin_amdgcn_wmma_f32_16x16x4_f32(
        false, A, false, B, (short)0, C, false, false);

    // D[j] at lane n<16 is rowsum[j]; at lane n>=16 it is rowsum[j+8].
    float half = D[0] + D[1] + D[2] + D[3] + D[4] + D[5] + D[6] + D[7];
    float other = __shfl_xor(half, 16, 32);       // wave32
    return half + other;
}

__global__ void rope_dot_partial(const float* __restrict__ vec,
                                 const float* __restrict__ rth,
                                 const int*   __restrict__ m_ptr,
                                 float*       __restrict__ partials,
                                 int n4) {
    const float mf = (float)(*m_ptr);
    const float4* __restrict__ v4 = (const float4*)vec;
    const float4* __restrict__ t4 = (const float4*)rth;

    float acc = 0.0f;
    const int stride = gridDim.x * blockDim.x;
    for (int g = blockIdx.x * blockDim.x + threadIdx.x; g < n4; g += stride) {
        float4 v = v4[g];   // (v0,v1,v2,v3): two pairs
        float4 t = t4[g];   // repeated thetas: t.x==t.y, t.z==t.w

        float e0 = cordic_est(mf * t.x);
        float e1 = cordic_est(mf * t.z);

        // e*(v0 + (-v1)) + e*(v1 + v0)  per pair (ests@vec + ests@neg_swapped)
        acc = fmaf(e0, (v.x - v.y) + (v.x + v.y), acc);
        acc = fmaf(e1, (v.z - v.w) + (v.z + v.w), acc);
    }

    float tot = block_reduce_wmma(acc);
    if (threadIdx.x == 0) partials[blockIdx.x] = tot;
}

__global__ void rope_dot_final(const float* __restrict__ partials,
                               int n,
                               float* __restrict__ out) {
    float acc = 0.0f;
    for (int i = threadIdx.x; i < n; i += blockDim.x) acc += partials[i];
    float tot = block_reduce_wmma(acc);
    if (threadIdx.x == 0) out[0] = tot;
}

extern "C" void kernel_launch(void* const* d_in, const int* in_sizes, int n_in,
                              void* d_out, int out_size, void* d_ws, size_t ws_size,
                              hipStream_t stream) {
    const float* vec = (const float*)d_in[0];   // float32 [16777216]
    const float* rth = (const float*)d_in[1];   // float32 [16777216]
    const int*   m   = (const int*)  d_in[2];   // scalar int (4096)
    float* out       = (float*)d_out;           // 1 element
    float* partials  = (float*)d_ws;

    const int L  = in_sizes[0];
    const int n4 = L / 4;                       // float4 groups (L is 16Mi)

    int blocks = 2048;                          // 2048*256 threads, 8 f4/thread
    size_t max_blocks = ws_size / sizeof(float);
    if ((size_t)blocks > max_blocks) blocks = (int)max_blocks;
    if (blocks < 1) blocks = 1;

    rope_dot_partial<<<blocks, 256, 0, stream>>>(vec, rth, m, partials, n4);
    rope_dot_final<<<1, 256, 0, stream>>>(partials, blocks, out);
}